// LinearAttention4_8589934686
// MI455X (gfx1250) — compile-verified
//
#include <hip/hip_runtime.h>
#include <hip/hip_bf16.h>

// ---------------------------------------------------------------------------
// LinearAttention (b=32, c=256, h=w=56) for MI455X / gfx1250, wave32.
// All GEMMs on V_WMMA_F32_16X16X4_F32 (fp32 matrix pipe, exact fp32 FMA).
// b128 global<->LDS staging, register double-buffering, split-K for kv,
// max-reuse block tiles (x read once per GEMM).
// ---------------------------------------------------------------------------

typedef __attribute__((ext_vector_type(2))) float v2f;
typedef __attribute__((ext_vector_type(8))) float v8f;

#define DIM    256
#define DHALF  128
#define HW     56
#define N_SP   3136            // 56*56
#define INV_N  (1.0f / 3136.0f)
#define KSPLIT 7               // kv split-K factor (98 chunks -> 14 each)
#define KV_ELEMS (32 * DHALF * DIM)   // 1,048,576 floats per kv slice

// LDS row strides (floats). All *4 bytes are multiples of 16 (b128-aligned).
// 36: 16 consecutive rows hit 16 distinct banks; row-banks ===0 (mod 4) so the
//     hi-half column offset (+2) is bank-disjoint. t*16*36 mod 64 == 0.
// 72: row+2 (hi half of wave) lands exactly +16 banks -> disjoint halves.
// 68: 16 consecutive rows hit 16 distinct banks (r*4 mod 64).
#define STR_A 36
#define STR_B 72
#define STR_T 68

__device__ __forceinline__ v8f wmma4(v2f a, v2f b, v8f c) {
  // D = A(16x4, f32) * B(4x16, f32) + C(16x16, f32)
  return __builtin_amdgcn_wmma_f32_16x16x4_f32(
      false, a, false, b, (short)0, c, false, false);
}

__device__ __forceinline__ float4 ldg4(const float* p) {
  return *reinterpret_cast<const float4*>(p);
}
__device__ __forceinline__ void sts4(float* p, float4 v) {
  *reinterpret_cast<float4*>(p) = v;
}

// ---------------------------------------------------------------------------
// Kernel 1: qk = elu(qk_w @ x + qk_b) + 1, split into q (o<128) / k (o>=128).
// Per batch GEMM: M=256 (o), N=3136 (n), K=256 (c).
// 256 thr (8 waves), block tile 256(all o) x 64 -> x is read exactly once.
// Wave w owns o-rows [32w, 32w+32): 2 m-subtiles x 4 n-subtiles.
// ---------------------------------------------------------------------------
__global__ __launch_bounds__(256) void qk_gemm_kernel(
    const float* __restrict__ x, const float* __restrict__ qk_w,
    const float* __restrict__ qk_b,
    float* __restrict__ q_ws, float* __restrict__ k_ws)
{
  __shared__ float As[256 * STR_A];  // 256 o x 32 k  (global layout)
  __shared__ float Bs[32 * STR_B];   // 32 k x 64 n   (global layout)

  const int b     = blockIdx.z;
  const int nBase = blockIdx.x * 64;
  const int tid   = threadIdx.x;
  const int wave  = tid >> 5;
  const int lane  = tid & 31;
  const int row16 = lane & 15;
  const int hi    = lane >> 4;
  const float* xb = x + (size_t)b * DIM * N_SP;

  v8f acc[8] = {};                   // [sub(2)][t(4)]
  float4 ra[8], rb[2];

  #pragma unroll
  for (int j = 0; j < 8; ++j) {      // prefetch chunk 0: A = qk_w 256x32
    const int f = tid + j * 256;
    ra[j] = ldg4(qk_w + (size_t)(f >> 3) * DIM + ((f & 7) * 4));
  }
  #pragma unroll
  for (int j = 0; j < 2; ++j) {      // B = x 32x64
    const int f = tid + j * 256;
    rb[j] = ldg4(xb + (size_t)(f >> 4) * N_SP + (nBase + (f & 15) * 4));
  }

  for (int kc = 0; kc < 8; ++kc) {
    #pragma unroll
    for (int j = 0; j < 8; ++j) {
      const int f = tid + j * 256;
      sts4(&As[(f >> 3) * STR_A + (f & 7) * 4], ra[j]);
    }
    #pragma unroll
    for (int j = 0; j < 2; ++j) {
      const int f = tid + j * 256;
      sts4(&Bs[(f >> 4) * STR_B + (f & 15) * 4], rb[j]);
    }
    __syncthreads();
    if (kc + 1 < 8) {
      const int kB = (kc + 1) * 32;
      #pragma unroll
      for (int j = 0; j < 8; ++j) {
        const int f = tid + j * 256;
        ra[j] = ldg4(qk_w + (size_t)(f >> 3) * DIM + (kB + (f & 7) * 4));
      }
      #pragma unroll
      for (int j = 0; j < 2; ++j) {
        const int f = tid + j * 256;
        rb[j] = ldg4(xb + (size_t)(kB + (f >> 4)) * N_SP + (nBase + (f & 15) * 4));
      }
    }
    #pragma unroll
    for (int kk = 0; kk < 32; kk += 4) {
      const int r0 = (wave * 32 + row16) * STR_A + kk + 2 * hi;
      v2f a0, a1;
      a0.x = As[r0];                a0.y = As[r0 + 1];
      a1.x = As[r0 + 16 * STR_A];   a1.y = As[r0 + 16 * STR_A + 1];
      #pragma unroll
      for (int t = 0; t < 4; ++t) {
        v2f bb;
        bb.x = Bs[(kk + 2 * hi) * STR_B + t * 16 + row16];
        bb.y = Bs[(kk + 2 * hi + 1) * STR_B + t * 16 + row16];
        acc[t]     = wmma4(a0, bb, acc[t]);
        acc[4 + t] = wmma4(a1, bb, acc[4 + t]);
      }
    }
    __syncthreads();
  }

  #pragma unroll
  for (int sub = 0; sub < 2; ++sub) {
    #pragma unroll
    for (int v = 0; v < 8; ++v) {
      const int o = wave * 32 + sub * 16 + v + 8 * hi;
      const float bias = qk_b[o];
      #pragma unroll
      for (int t = 0; t < 4; ++t) {
        const int n = nBase + t * 16 + row16;
        float z = acc[sub * 4 + t][v] + bias;
        float val = (z > 0.0f) ? (z + 1.0f) : __expf(z);
        if (o < DHALF)
          q_ws[((size_t)b * DHALF + o) * N_SP + n] = val;
        else
          k_ws[((size_t)b * DHALF + (o - DHALF)) * N_SP + n] = val;
      }
    }
  }
}

// ---------------------------------------------------------------------------
// Kernel 2: kv_part[s][b,d,c] = sum_{n in split s} k[b,d,n] * x[b,c,n].
// GEMM M=128 (d), N=256 (c), K=3136 (n) split KSPLIT ways (14 chunks each).
// 256 thr, block tile 128(all d) x 128(c) -> x read once, k read twice.
// Wave w owns d-rows [16w,16w+16) x 128 c (8 C tiles).
// ---------------------------------------------------------------------------
__global__ __launch_bounds__(256) void kv_gemm_kernel(
    const float* __restrict__ x, const float* __restrict__ k_ws,
    float* __restrict__ kv_part)
{
  __shared__ float As[128 * STR_A];  // 128 d x 32 n
  __shared__ float Bs[128 * STR_A];  // 128 c x 32 n

  const int b     = blockIdx.z / KSPLIT;
  const int seg   = blockIdx.z % KSPLIT;
  const int cBase = blockIdx.x * 128;
  const int kOrig = seg * (N_SP / KSPLIT);     // 448 n per split
  const int tid   = threadIdx.x;
  const int wave  = tid >> 5;
  const int lane  = tid & 31;
  const int row16 = lane & 15;
  const int hi    = lane >> 4;
  const float* kb = k_ws + (size_t)b * DHALF * N_SP;
  const float* xb = x    + (size_t)b * DIM   * N_SP;

  v8f acc[8] = {};                   // 8 c-subtiles
  float4 ra[4], rb[4];

  #pragma unroll
  for (int j = 0; j < 4; ++j) {
    const int f = tid + j * 256;
    ra[j] = ldg4(kb + (size_t)(f >> 3) * N_SP + (kOrig + (f & 7) * 4));
    rb[j] = ldg4(xb + (size_t)(cBase + (f >> 3)) * N_SP + (kOrig + (f & 7) * 4));
  }

  for (int kc = 0; kc < 14; ++kc) {
    #pragma unroll
    for (int j = 0; j < 4; ++j) {
      const int f = tid + j * 256;
      sts4(&As[(f >> 3) * STR_A + (f & 7) * 4], ra[j]);
      sts4(&Bs[(f >> 3) * STR_A + (f & 7) * 4], rb[j]);
    }
    __syncthreads();
    if (kc + 1 < 14) {
      const int kB = kOrig + (kc + 1) * 32;
      #pragma unroll
      for (int j = 0; j < 4; ++j) {
        const int f = tid + j * 256;
        ra[j] = ldg4(kb + (size_t)(f >> 3) * N_SP + (kB + (f & 7) * 4));
        rb[j] = ldg4(xb + (size_t)(cBase + (f >> 3)) * N_SP + (kB + (f & 7) * 4));
      }
    }
    #pragma unroll
    for (int kk = 0; kk < 32; kk += 4) {
      const int r0 = (wave * 16 + row16) * STR_A + kk + 2 * hi;
      v2f a;
      a.x = As[r0]; a.y = As[r0 + 1];
      #pragma unroll
      for (int t = 0; t < 8; ++t) {
        v2f bb;   // B[k=n][col=c] = x[c][n] -> gather from [c][n] tile
        bb.x = Bs[(t * 16 + row16) * STR_A + kk + 2 * hi];
        bb.y = Bs[(t * 16 + row16) * STR_A + kk + 2 * hi + 1];
        acc[t] = wmma4(a, bb, acc[t]);
      }
    }
    __syncthreads();
  }

  float* dst = kv_part + (size_t)seg * KV_ELEMS;
  #pragma unroll
  for (int v = 0; v < 8; ++v) {
    const int d = wave * 16 + v + 8 * hi;
    #pragma unroll
    for (int t = 0; t < 8; ++t) {
      const int c = cBase + t * 16 + row16;
      dst[((size_t)b * DHALF + d) * DIM + c] = acc[t][v];
    }
  }
}

// ---------------------------------------------------------------------------
// Kernel 2r: kv = (1/n) * sum_s kv_part[s]   (float4, deterministic)
// ---------------------------------------------------------------------------
__global__ __launch_bounds__(256) void kv_reduce_kernel(
    const float* __restrict__ kv_part, float* __restrict__ kv_ws)
{
  const int i4 = blockIdx.x * 256 + threadIdx.x;   // over KV_ELEMS/4 = 262144
  const float4* p = reinterpret_cast<const float4*>(kv_part);
  float4 s = p[i4];
  #pragma unroll
  for (int sg = 1; sg < KSPLIT; ++sg) {
    float4 v = p[(size_t)sg * (KV_ELEMS / 4) + i4];
    s.x += v.x; s.y += v.y; s.z += v.z; s.w += v.w;
  }
  s.x *= INV_N; s.y *= INV_N; s.z *= INV_N; s.w *= INV_N;
  reinterpret_cast<float4*>(kv_ws)[i4] = s;
}

// ---------------------------------------------------------------------------
// Kernel 2b: kmean[b,d] = mean_n k[b,d,n]. One block per (b,d).
// ---------------------------------------------------------------------------
__global__ __launch_bounds__(128) void kmean_kernel(
    const float* __restrict__ k_ws, float* __restrict__ kmean)
{
  __shared__ float red[128];
  const float4* kp =
      reinterpret_cast<const float4*>(k_ws + (size_t)blockIdx.x * N_SP);
  float s = 0.0f;
  for (int n4 = threadIdx.x; n4 < N_SP / 4; n4 += 128) {
    float4 v = kp[n4];
    s += (v.x + v.y) + (v.z + v.w);
  }
  red[threadIdx.x] = s;
  __syncthreads();
  #pragma unroll
  for (int off = 64; off > 0; off >>= 1) {
    if (threadIdx.x < off) red[threadIdx.x] += red[threadIdx.x + off];
    __syncthreads();
  }
  if (threadIdx.x == 0) kmean[blockIdx.x] = red[0] * INV_N;
}

// ---------------------------------------------------------------------------
// Kernel PE: depthwise 3x3 SAME conv (cross-correlation) + pe_b -> writes out.
// ---------------------------------------------------------------------------
__global__ __launch_bounds__(256) void pe_kernel(
    const float* __restrict__ x, const float* __restrict__ pe_w,
    const float* __restrict__ pe_b, float* __restrict__ out)
{
  const size_t idx = (size_t)blockIdx.x * 256 + threadIdx.x;
  const int xx = (int)(idx % HW);
  size_t t = idx / HW;
  const int y = (int)(t % HW); t /= HW;
  const int c = (int)(t % DIM);
  const int b = (int)(t / DIM);

  const float* xc = x + ((size_t)b * DIM + c) * N_SP;
  const float* wc = pe_w + (size_t)c * 9;
  float acc = pe_b[c];
  #pragma unroll
  for (int ky = 0; ky < 3; ++ky) {
    const int yy = y + ky - 1;
    if (yy < 0 || yy >= HW) continue;
    #pragma unroll
    for (int kx = 0; kx < 3; ++kx) {
      const int xs = xx + kx - 1;
      if (xs < 0 || xs >= HW) continue;
      acc += wc[ky * 3 + kx] * xc[yy * HW + xs];
    }
  }
  out[idx] = acc;
}

// ---------------------------------------------------------------------------
// Kernel 3: num = q^T @ kv, den = q^T @ kmean; out += num/den.
// GEMM M=3136 (n), N=256 (c), K=128 (d) fully LDS-resident; den fused;
// result staged [c][n] in LDS so the out+= RMW is a coalesced b128 along n.
// q/kv re-reads across c-tiles are L2-resident (q[b]=1.6MB, kv[b]=131KB).
// ---------------------------------------------------------------------------
__global__ __launch_bounds__(128) void attn_out_kernel(
    const float* __restrict__ q_ws, const float* __restrict__ kv_ws,
    const float* __restrict__ kmean, float* __restrict__ out)
{
  __shared__ float Qs[DHALF * STR_B];  // 128 d x 64 n  (global layout)
  __shared__ float Bs[DHALF * STR_B];  // 128 d x 64 c  (global layout)
  __shared__ float Ts[64 * STR_T];     // 64 c x 64 n   result tile
  __shared__ float kms[DHALF];
  __shared__ float dens[64];

  const int b     = blockIdx.z;
  const int nBase = blockIdx.y * 64;
  const int cBase = blockIdx.x * 64;
  const int tid   = threadIdx.x;
  const int wave  = tid >> 5;
  const int lane  = tid & 31;
  const int row16 = lane & 15;
  const int hi    = lane >> 4;
  const float* qb  = q_ws  + (size_t)b * DHALF * N_SP;
  const float* kvb = kv_ws + (size_t)b * DHALF * DIM;

  #pragma unroll
  for (int i = 0; i < 16; ++i) {       // both tiles: b128 load + b128 store
    const int f = tid + i * 128;
    const int d = f >> 4, e4 = (f & 15) * 4;
    sts4(&Qs[d * STR_B + e4], ldg4(qb + (size_t)d * N_SP + nBase + e4));
    sts4(&Bs[d * STR_B + e4], ldg4(kvb + (size_t)d * DIM + cBase + e4));
  }
  if (tid < DHALF) kms[tid] = kmean[b * DHALF + tid];
  __syncthreads();

  if (tid < 64) {                      // fused denominator for this n-tile
    float s = 1e-6f;
    for (int d = 0; d < DHALF; ++d) s += Qs[d * STR_B + tid] * kms[d];
    dens[tid] = s;
  }
  __syncthreads();

  v8f acc[4] = {};
  for (int kk = 0; kk < DHALF; kk += 4) {
    v2f a;   // A[m=n][k=d] = q[d][n] -> gather from [d][n] tile
    a.x = Qs[(kk + 2 * hi) * STR_B + wave * 16 + row16];
    a.y = Qs[(kk + 2 * hi + 1) * STR_B + wave * 16 + row16];
    #pragma unroll
    for (int t = 0; t < 4; ++t) {
      v2f bb;
      bb.x = Bs[(kk + 2 * hi) * STR_B + t * 16 + row16];
      bb.y = Bs[(kk + 2 * hi + 1) * STR_B + t * 16 + row16];
      acc[t] = wmma4(a, bb, acc[t]);
    }
  }

  #pragma unroll
  for (int v = 0; v < 8; ++v) {        // divide by den, stage as [c][n]
    const int nLocal = wave * 16 + v + 8 * hi;
    #pragma unroll
    for (int t = 0; t < 4; ++t) {
      Ts[(t * 16 + row16) * STR_T + nLocal] = acc[t][v] / dens[nLocal];
    }
  }
  __syncthreads();

  #pragma unroll
  for (int i = 0; i < 8; ++i) {        // out += attn, b128 RMW along n
    const int f  = tid + i * 128;      // 64 c rows x 16 float4 per row
    const int cr = f >> 4, n4 = (f & 15) * 4;
    float* op = out + ((size_t)b * DIM + cBase + cr) * N_SP + nBase + n4;
    float4 vo = ldg4(op);
    float4 tv = ldg4(&Ts[cr * STR_T + n4]);
    vo.x += tv.x; vo.y += tv.y; vo.z += tv.z; vo.w += tv.w;
    sts4(op, vo);
  }
}

// ---------------------------------------------------------------------------
// Workspace layout (bytes):
//   q_ws    @ 0            : 51,380,224
//   k_ws    @ 51,380,224   : 51,380,224
//   kv_ws   @ 102,760,448  :  4,194,304
//   kmean   @ 106,954,752  :     16,384
//   kv_part @ 106,971,136  : 29,360,128  (7 split-K slices)
// Total ~130 MB.
// ---------------------------------------------------------------------------
extern "C" void kernel_launch(void* const* d_in, const int* in_sizes, int n_in,
                              void* d_out, int out_size, void* d_ws, size_t ws_size,
                              hipStream_t stream) {
  const float* x    = (const float*)d_in[0];
  const float* qk_w = (const float*)d_in[1];
  const float* qk_b = (const float*)d_in[2];
  const float* pe_w = (const float*)d_in[3];
  const float* pe_b = (const float*)d_in[4];
  float* out = (float*)d_out;

  char* ws = (char*)d_ws;
  float* q_ws    = (float*)(ws);
  float* k_ws    = (float*)(ws + 51380224);
  float* kv_ws   = (float*)(ws + 102760448);
  float* kmean   = (float*)(ws + 106954752);
  float* kv_part = (float*)(ws + 106971136);

  // 1) qk GEMM + elu+1 -> q_ws, k_ws   (x read exactly once)
  qk_gemm_kernel<<<dim3(49, 1, 32), 256, 0, stream>>>(x, qk_w, qk_b, q_ws, k_ws);
  // 2) kv split-K partials, then deterministic reduce (* 1/n)
  kv_gemm_kernel<<<dim3(2, 1, 32 * KSPLIT), 256, 0, stream>>>(x, k_ws, kv_part);
  kv_reduce_kernel<<<dim3(KV_ELEMS / 4 / 256), 256, 0, stream>>>(kv_part, kv_ws);
  // 2b) kmean
  kmean_kernel<<<dim3(32 * DHALF), 128, 0, stream>>>(k_ws, kmean);
  // 3) out = pe + pe_b (initializes output)
  pe_kernel<<<dim3((32u * DIM * N_SP) / 256), 256, 0, stream>>>(x, pe_w, pe_b, out);
  // 4) out += (q^T kv) / (q^T kmean + 1e-6)
  attn_out_kernel<<<dim3(4, 49, 32), 128, 0, stream>>>(q_ws, kv_ws, kmean, out);
}